// KSparseLinear_87058987090491
// MI455X (gfx1250) — compile-verified
//
#include <hip/hip_runtime.h>

typedef __attribute__((ext_vector_type(2))) float v2f;
typedef __attribute__((ext_vector_type(8))) float v8f;

#define IN_F   4096
#define OUT_F  8
#define KSEL   64
#define TILE_M 16

// ---------------------------------------------------------------------------
// Kernel 1: deterministic top-64 selection by rank.
// rank(j) = #{ i : |fi[i]| > |fi[j]|  ||  (|fi[i]| == |fi[j]| && i < j) }
// Ranks are a permutation of 0..4095, so ranks 0..63 each occur exactly once
// -> collision-free ordered write of the compacted index list.
// ---------------------------------------------------------------------------
__global__ __launch_bounds__(256) void k_rank_select(const float* __restrict__ fi,
                                                     int* __restrict__ idx_ws) {
    __shared__ float sfi[IN_F];
    const int t = threadIdx.x;
    for (int j = t; j < IN_F; j += 256) sfi[j] = fabsf(fi[j]);
    __syncthreads();

    const int j = blockIdx.x * 256 + t;       // grid = 16 blocks -> j in [0,4096)
    const float v = sfi[j];
    int rank = 0;
    for (int i = 0; i < IN_F; ++i) {          // LDS broadcast read per iter
        const float u = sfi[i];
        rank += (u > v || (u == v && i < j)) ? 1 : 0;
    }
    if (rank < KSEL) idx_ws[rank] = j;
}

// ---------------------------------------------------------------------------
// Kernel 2: pack W[:, idx] into a 64x16 fp32 tile (B-matrix), N=8..15 zero.
// wp[k*16 + n] = (n < 8) ? weight[n*IN_F + idx[k]] : 0
// ---------------------------------------------------------------------------
__global__ __launch_bounds__(256) void k_pack_w(const float* __restrict__ weight,
                                                const int* __restrict__ idx_ws,
                                                float* __restrict__ wp_ws) {
    for (unsigned e = threadIdx.x; e < KSEL * 16u; e += 256u) {
        const unsigned k = e >> 4;
        const unsigned n = e & 15u;
        wp_ws[e] = (n < OUT_F) ? weight[n * (unsigned)IN_F + (unsigned)idx_ws[k]]
                               : 0.0f;
    }
}

// ---------------------------------------------------------------------------
// Kernel 3: per 16-row tile, gather the 64 selected columns and run a
// 16x16x64 fp32 matmul as 16 chained V_WMMA_F32_16X16X4_F32 ops.
//
// VGPR layouts (CDNA5 ISA 7.12.2, wave32):
//   A 16x4 f32 : lane L: M = L&15 ; v0 holds K = 2*(L>>4), v1 holds K+1
//   B 4x16 f32 : lane L: N = L&15 ; v0 holds K = 2*(L>>4), v1 holds K+1
//   C/D 16x16  : VGPR v: M = v (lanes 0-15), M = v+8 (lanes 16-31); N = L&15
//
// All global accesses use 32-bit unsigned element offsets against uniform
// (SGPR) base pointers so the backend emits single-instruction GVS loads
// (saddr + voffset, scale_offset) with no per-gather 64-bit address math.
// Max x offset = (65535*4096 + 4095) < 2^28, fits the signed-i32 VADDR.
// ---------------------------------------------------------------------------
__global__ __launch_bounds__(256) void k_gemm(const float* __restrict__ x,
                                              const float* __restrict__ bias,
                                              const int* __restrict__ idx_ws,
                                              const float* __restrict__ wp_ws,
                                              float* __restrict__ out) {
    const unsigned lane = threadIdx.x & 31u;
    const unsigned wave = threadIdx.x >> 5;          // 8 waves / block
    const unsigned tile = blockIdx.x * 8u + wave;    // 4096 tiles total
    const unsigned rowBase = tile * TILE_M;
    const unsigned half = lane >> 4;                 // 0: K pair {0,1}, 1: {2,3}
    const unsigned nl   = lane & 15u;                // M for A, N for B/D

    // ---- preload B tile (64x16) into registers in WMMA layout --------------
    v2f breg[16];
#pragma unroll
    for (int c = 0; c < 16; ++c) {
        const unsigned k0 = 4u * c + 2u * half;
        breg[c].x = wp_ws[k0 * 16u + nl];
        breg[c].y = wp_ws[(k0 + 1u) * 16u + nl];
    }

    // ---- precompute 32-bit element offsets for the gathers -----------------
    const unsigned rowElem = (rowBase + nl) * (unsigned)IN_F;   // < 2^28
    unsigned offA0[16], offA1[16];
#pragma unroll
    for (int c = 0; c < 16; ++c) {
        const unsigned k0 = 4u * c + 2u * half;
        offA0[c] = rowElem + (unsigned)idx_ws[k0];
        offA1[c] = rowElem + (unsigned)idx_ws[k0 + 1u];
    }

    // ---- issue all 32 gathers before the WMMA chain (loads in flight) ------
    v2f areg[16];
#pragma unroll
    for (int c = 0; c < 16; ++c) {
        areg[c].x = x[offA0[c]];
        areg[c].y = x[offA1[c]];
    }

    // ---- 16x16x64 fp32 matmul: 16 chained K=4 WMMA steps -------------------
    v8f acc = {};
#pragma unroll
    for (int c = 0; c < 16; ++c) {
        acc = __builtin_amdgcn_wmma_f32_16x16x4_f32(
            /*neg_a=*/false, areg[c],
            /*neg_b=*/false, breg[c],
            /*c_mod=*/(short)0, acc,
            /*reuse_a=*/false, /*reuse_b=*/false);
    }

    // ---- store D: only N < 8 columns are real outputs ----------------------
    if (nl < OUT_F) {
        const float bval = bias[nl];
#pragma unroll
        for (int v = 0; v < 8; ++v) {
            const unsigned row = rowBase + (unsigned)v + half * 8u;  // M index
            out[row * (unsigned)OUT_F + nl] = acc[v] + bval;
        }
    }
}

// ---------------------------------------------------------------------------
// d_in order: x[65536*4096] f32, weight[8*4096] f32, bias[8] f32,
//             feature_importance[4096] f32, k (==64, hardcoded)
// d_ws layout: [0,256)   int idx_ws[64]
//              [256, +4KB) float wp_ws[64*16]
// ---------------------------------------------------------------------------
extern "C" void kernel_launch(void* const* d_in, const int* in_sizes, int n_in,
                              void* d_out, int out_size, void* d_ws, size_t ws_size,
                              hipStream_t stream) {
    const float* x      = (const float*)d_in[0];
    const float* weight = (const float*)d_in[1];
    const float* bias   = (const float*)d_in[2];
    const float* fi     = (const float*)d_in[3];
    (void)in_sizes; (void)n_in; (void)out_size; (void)ws_size;

    int*   idx_ws = (int*)d_ws;
    float* wp_ws  = (float*)((char*)d_ws + 256);
    float* out    = (float*)d_out;

    k_rank_select<<<16, 256, 0, stream>>>(fi, idx_ws);
    k_pack_w<<<1, 256, 0, stream>>>(weight, idx_ws, wp_ws);
    k_gemm<<<512, 256, 0, stream>>>(x, bias, idx_ws, wp_ws, out);
}